// E3_transformer_test_42528766165478
// MI455X (gfx1250) — compile-verified
//
#include <hip/hip_runtime.h>
#include <hip/hip_bf16.h>
#include <math.h>

// ---------------------------------------------------------------------------
// Invariant Point Attention for MI455X (gfx1250, wave32, WMMA).
// Bandwidth-bound: z = 302 MB dominates; fuse z@Wb and z@Wdz into one pass
// (3 f32 WMMA accumulators share each z fragment). All GEMMs use
// V_WMMA_F32_16X16X4_F32 (fp32 in/out -> full precision, compute is free
// relative to the 23.3 TB/s HBM floor of ~13 us for the z stream).
// Workspace requirement: ~145 MB (pair_z 75.5 MB + b/att 28.3 MB each + misc).
// ---------------------------------------------------------------------------

#define N_  768
#define CS_ 384
#define CZ_ 128
#define CH_ 16
#define H_  12
#define PQ_ 4
#define PV_ 8

typedef __attribute__((ext_vector_type(2))) float v2f;
typedef __attribute__((ext_vector_type(8))) float v8f;

static __device__ __forceinline__ v8f wmma4(v2f a, v2f b, v8f c) {
  // D = A(16x4) * B(4x16) + C, fp32
  return __builtin_amdgcn_wmma_f32_16x16x4_f32(false, a, false, b, (short)0, c,
                                               false, false);
}

// ---------------------------------------------------------------------------
// Generic Out[M][Nout] = A[M][K] @ W[Nout][K]^T + bias (+resid).  1 wave/block,
// block computes one 16x16 tile. lda == K for all uses here; ldw == K.
// ---------------------------------------------------------------------------
__global__ void gemm_awt_kernel(const float* __restrict__ A,
                                const float* __restrict__ W,
                                const float* __restrict__ bias,
                                const float* __restrict__ resid,
                                float* __restrict__ Out,
                                int K, int Nout, int ldo) {
  const int lane = threadIdx.x;
  const int half = lane >> 4, idx = lane & 15;
  const int m0 = blockIdx.x * 16, n0 = blockIdx.y * 16;
  const int col  = n0 + idx;
  const int wcol = (col < Nout) ? col : (Nout - 1);
  const float* arow = A + (size_t)(m0 + idx) * K + 2 * half;
  const float* wrow = W + (size_t)wcol * K + 2 * half;
  const float binit = bias ? bias[wcol] : 0.0f;
  v8f c;
#pragma unroll
  for (int r = 0; r < 8; ++r) c[r] = binit;
  for (int k = 0; k < K; k += 4) {
    v2f a = *(const v2f*)(arow + k);
    v2f b = *(const v2f*)(wrow + k);
    c = wmma4(a, b, c);
  }
  if (col < Nout) {
#pragma unroll
    for (int r = 0; r < 8; ++r) {
      const int row = m0 + r + 8 * half;
      float v = c[r];
      if (resid) v += resid[(size_t)row * ldo + col];
      Out[(size_t)row * ldo + col] = v;
    }
  }
}

// ---------------------------------------------------------------------------
// Rotate + translate query / key-value points (split3 layout from reference).
// ---------------------------------------------------------------------------
__global__ void pts_kernel(const float* __restrict__ qp,
                           const float* __restrict__ kvp,
                           const float* __restrict__ rot,
                           const float* __restrict__ trans,
                           float* __restrict__ q_pts,
                           float* __restrict__ kv_pts) {
  int tid = blockIdx.x * blockDim.x + threadIdx.x;
  if (tid >= N_ * 192) return;
  int n = tid / 192, t = tid % 192;
  const float* R  = rot + (size_t)n * 9;
  const float* tr = trans + (size_t)n * 3;
  float p0, p1, p2;
  float* dst;
  if (t < 48) {                       // H*PQ = 48 query points
    const float* src = qp + (size_t)n * 144;
    p0 = src[t]; p1 = src[48 + t]; p2 = src[96 + t];
    dst = q_pts + ((size_t)n * 48 + t) * 3;
  } else {                            // H*(PQ+PV) = 144 kv points
    int p = t - 48;
    const float* src = kvp + (size_t)n * 432;
    p0 = src[p]; p1 = src[144 + p]; p2 = src[288 + p];
    dst = kv_pts + ((size_t)n * 144 + p) * 3;
  }
  dst[0] = R[0] * p0 + R[1] * p1 + R[2] * p2 + tr[0];
  dst[1] = R[3] * p0 + R[4] * p1 + R[5] * p2 + tr[1];
  dst[2] = R[6] * p0 + R[7] * p1 + R[8] * p2 + tr[2];
}

// ---------------------------------------------------------------------------
// Pack per-head fragments:
//   qpk/kpk[n][h][28] = [s1*q (16) | pts (12)]  (s1 only on Q side)
//   mQ/mK[n][h]       = -0.5 * |pts|^2
//   vpk[n][h][40]     = [v (16) | v_pts flattened (24)]
// ---------------------------------------------------------------------------
__global__ void pack_kernel(const float* __restrict__ q,
                            const float* __restrict__ kv,
                            const float* __restrict__ q_pts,
                            const float* __restrict__ kv_pts,
                            float* __restrict__ qpk, float* __restrict__ kpk,
                            float* __restrict__ mQ, float* __restrict__ mK,
                            float* __restrict__ vpk) {
  int tid = blockIdx.x * blockDim.x + threadIdx.x;
  if (tid >= N_ * H_ * 3) return;
  int which = tid % 3;
  int nh = tid / 3;
  int h = nh % H_, n = nh / H_;
  const float s1 = rsqrtf(3.0f * CH_);
  if (which == 0) {
    float* dst = qpk + ((size_t)n * H_ + h) * 28;
    const float* qs = q + (size_t)n * (H_ * CH_) + h * CH_;
    for (int kk = 0; kk < 16; ++kk) dst[kk] = s1 * qs[kk];
    const float* ps = q_pts + ((size_t)n * 48 + h * PQ_) * 3;
    float m = 0.f;
    for (int t = 0; t < 12; ++t) { float v = ps[t]; dst[16 + t] = v; m += v * v; }
    mQ[(size_t)n * H_ + h] = -0.5f * m;
  } else if (which == 1) {
    float* dst = kpk + ((size_t)n * H_ + h) * 28;
    const float* ks = kv + (size_t)n * (2 * H_ * CH_) + h * (2 * CH_);
    for (int kk = 0; kk < 16; ++kk) dst[kk] = ks[kk];
    const float* ps = kv_pts + ((size_t)n * 144 + h * 12) * 3;   // k_pts: pts 0..3
    float m = 0.f;
    for (int t = 0; t < 12; ++t) { float v = ps[t]; dst[16 + t] = v; m += v * v; }
    mK[(size_t)n * H_ + h] = -0.5f * m;
  } else {
    float* dst = vpk + ((size_t)n * H_ + h) * 40;
    const float* vs = kv + (size_t)n * (2 * H_ * CH_) + h * (2 * CH_) + CH_;
    for (int kk = 0; kk < 16; ++kk) dst[kk] = vs[kk];
    const float* ps = kv_pts + ((size_t)n * 144 + h * 12 + PQ_) * 3; // v_pts: pts 4..11
    for (int t = 0; t < 24; ++t) dst[16 + t] = ps[t];
  }
}

// ---------------------------------------------------------------------------
// Fused z GEMM: reads each z tile ONCE, produces b (12 cols, padded tile) and
// pair_z (32 cols) with 3 WMMA accumulators.  M = N*N rows, K = 128.
// ---------------------------------------------------------------------------
__global__ void bz_kernel(const float* __restrict__ z,
                          const float* __restrict__ Wb, const float* __restrict__ bb,
                          const float* __restrict__ Wdz, const float* __restrict__ bdz,
                          float* __restrict__ b_out, float* __restrict__ pz_out) {
  const int lane = threadIdx.x;
  const int half = lane >> 4, idx = lane & 15;
  const size_t m0 = (size_t)blockIdx.x * 16;
  const float* arow = z + (m0 + idx) * CZ_ + 2 * half;
  const int hcol = (idx < H_) ? idx : (H_ - 1);
  const float* w0 = Wb  + (size_t)hcol * CZ_ + 2 * half;
  const float* w1 = Wdz + (size_t)idx * CZ_ + 2 * half;
  const float* w2 = Wdz + (size_t)(16 + idx) * CZ_ + 2 * half;
  v8f c0, c1, c2;
  const float i0 = bb[hcol], i1 = bdz[idx], i2 = bdz[16 + idx];
#pragma unroll
  for (int r = 0; r < 8; ++r) { c0[r] = i0; c1[r] = i1; c2[r] = i2; }
  // Prefetch next tile's rows of the z stream (speculative; dropped if OOB).
  __builtin_prefetch(z + (m0 + 16 + idx) * CZ_ + 2 * half, 0, 0);
  for (int k = 0; k < CZ_; k += 4) {
    v2f a = *(const v2f*)(arow + k);
    c0 = wmma4(a, *(const v2f*)(w0 + k), c0);
    c1 = wmma4(a, *(const v2f*)(w1 + k), c1);
    c2 = wmma4(a, *(const v2f*)(w2 + k), c2);
  }
#pragma unroll
  for (int r = 0; r < 8; ++r) {
    const size_t row = m0 + r + 8 * half;
    if (idx < H_) b_out[row * H_ + idx] = c0[r];
    pz_out[row * 32 + idx]      = c1[r];
    pz_out[row * 32 + 16 + idx] = c2[r];
  }
}

// ---------------------------------------------------------------------------
// Attention logits per head: K=28 WMMA GEMM over packed fragments, accumulator
// seeded with s2*b + (-0.5|q_pts|^2) + (-0.5|k_pts|^2) + mask bias.
// ---------------------------------------------------------------------------
__global__ void logits_kernel(const float* __restrict__ qpk,
                              const float* __restrict__ kpk,
                              const float* __restrict__ mQ,
                              const float* __restrict__ mK,
                              const float* __restrict__ bmat,
                              const float* __restrict__ mask,
                              float* __restrict__ att) {
  const int lane = threadIdx.x;
  const int half = lane >> 4, idx = lane & 15;
  const int h = blockIdx.z;
  const int m0 = blockIdx.x * 16, n0 = blockIdx.y * 16;
  const int col = n0 + idx;
  const float s2 = 0.57735026918962576f;  // sqrt(1/3)
  const float* arow = qpk + ((size_t)(m0 + idx) * H_ + h) * 28 + 2 * half;
  const float* brow = kpk + ((size_t)col * H_ + h) * 28 + 2 * half;
  v8f c;
#pragma unroll
  for (int r = 0; r < 8; ++r) {
    const int row = m0 + r + 8 * half;
    c[r] = s2 * bmat[((size_t)row * N_ + col) * H_ + h]
         + mQ[(size_t)row * H_ + h] + mK[(size_t)col * H_ + h]
         + 100000.0f * (mask[row] * mask[col] - 1.0f);
  }
  for (int k = 0; k < 28; k += 4) {
    v2f a = *(const v2f*)(arow + k);
    v2f b = *(const v2f*)(brow + k);
    c = wmma4(a, b, c);
  }
#pragma unroll
  for (int r = 0; r < 8; ++r) {
    const int row = m0 + r + 8 * half;
    att[((size_t)h * N_ + row) * N_ + col] = c[r];
  }
}

// ---------------------------------------------------------------------------
// In-place row softmax, one wave per (h,i) row.
// ---------------------------------------------------------------------------
__global__ void softmax_kernel(float* __restrict__ att) {
  const int lane = threadIdx.x;
  float* row = att + (size_t)blockIdx.x * N_;
  float m = -3.0e38f;
  for (int j = lane; j < N_; j += 32) m = fmaxf(m, row[j]);
#pragma unroll
  for (int o = 16; o > 0; o >>= 1) m = fmaxf(m, __shfl_xor(m, o, 32));
  float s = 0.f;
  for (int j = lane; j < N_; j += 32) {
    float e = __expf(row[j] - m);
    row[j] = e;
    s += e;
  }
#pragma unroll
  for (int o = 16; o > 0; o >>= 1) s += __shfl_xor(s, o, 32);
  const float inv = 1.0f / s;
  for (int j = lane; j < N_; j += 32) row[j] *= inv;
}

// ---------------------------------------------------------------------------
// o and o_pt: per head, Out[i][0..39] = sum_j att[h][i][j] * vpk[j][h][0..39].
// ---------------------------------------------------------------------------
__global__ void ov_kernel(const float* __restrict__ att,
                          const float* __restrict__ vpk,
                          float* __restrict__ feats, float* __restrict__ o_pt) {
  const int lane = threadIdx.x;
  const int half = lane >> 4, idx = lane & 15;
  const int h = blockIdx.z;
  const int m0 = blockIdx.x * 16;
  const int nloc = blockIdx.y * 16 + idx;   // 0..47 (40 valid)
  const int ncl = (nloc < 40) ? nloc : 39;
  const float* arow = att + ((size_t)h * N_ + m0 + idx) * N_ + 2 * half;
  v8f c;
#pragma unroll
  for (int r = 0; r < 8; ++r) c[r] = 0.0f;
  for (int k = 0; k < N_; k += 4) {
    v2f a = *(const v2f*)(arow + k);
    const size_t j = (size_t)k + 2 * half;
    v2f b;
    b.x = vpk[(j * H_ + h) * 40 + ncl];
    b.y = vpk[((j + 1) * H_ + h) * 40 + ncl];
    c = wmma4(a, b, c);
  }
#pragma unroll
  for (int r = 0; r < 8; ++r) {
    const int row = m0 + r + 8 * half;
    if (nloc < 16) {
      feats[(size_t)row * 960 + h * CH_ + nloc] = c[r];            // o
    } else if (nloc < 40) {
      const int t = nloc - 16;                                     // o_pt (global frame)
      o_pt[((size_t)row * 96 + h * PV_ + t / 3) * 3 + (t % 3)] = c[r];
    }
  }
}

// ---------------------------------------------------------------------------
// o_pair: per i, Out[h][d] = sum_j att[h][i][j] * pair_z[i][j][d]. M=12 (pad 16).
// ---------------------------------------------------------------------------
__global__ void opair_kernel(const float* __restrict__ att,
                             const float* __restrict__ pz,
                             float* __restrict__ feats) {
  const int lane = threadIdx.x;
  const int half = lane >> 4, idx = lane & 15;
  const int i = blockIdx.x;
  const int d = blockIdx.y * 16 + idx;      // 0..31
  const int hcl = (idx < H_) ? idx : (H_ - 1);
  const float* arow  = att + ((size_t)hcl * N_ + i) * N_ + 2 * half;
  const float* pzrow = pz + (size_t)i * N_ * 32 + d;
  v8f c;
#pragma unroll
  for (int r = 0; r < 8; ++r) c[r] = 0.0f;
  for (int k = 0; k < N_; k += 4) {
    v2f a = *(const v2f*)(arow + k);
    const size_t j = (size_t)k + 2 * half;
    v2f b;
    b.x = pzrow[j * 32];
    b.y = pzrow[(j + 1) * 32];
    c = wmma4(a, b, c);
  }
#pragma unroll
  for (int r = 0; r < 8; ++r) {
    const int hrow = r + 8 * half;
    if (hrow < H_) feats[(size_t)i * 960 + 576 + hrow * 32 + d] = c[r];
  }
}

// ---------------------------------------------------------------------------
// Inverse-rotate o_pt, dists, and l1_out.  Also fills feats[192..575].
// ---------------------------------------------------------------------------
__global__ void post_kernel(const float* __restrict__ o_pt,
                            const float* __restrict__ rot,
                            const float* __restrict__ trans,
                            const float* __restrict__ l1_feats,
                            const float* __restrict__ Wl1,
                            float* __restrict__ feats,
                            float* __restrict__ l1_out) {
  const int n = blockIdx.x;
  const int t = threadIdx.x;
  const float* R  = rot + (size_t)n * 9;
  const float* tr = trans + (size_t)n * 3;
  if (t < 96) {
    const float* g = o_pt + ((size_t)n * 96 + t) * 3;
    const float d0 = g[0] - tr[0], d1 = g[1] - tr[1], d2 = g[2] - tr[2];
    const float l0 = R[0] * d0 + R[3] * d1 + R[6] * d2;   // R^T * d
    const float l1 = R[1] * d0 + R[4] * d1 + R[7] * d2;
    const float l2 = R[2] * d0 + R[5] * d1 + R[8] * d2;
    float* f = feats + (size_t)n * 960;
    f[192 + t] = l0;
    f[288 + t] = l1;
    f[384 + t] = l2;
    f[480 + t] = sqrtf(l0 * l0 + l1 * l1 + l2 * l2 + 1e-8f);
  } else if (t < 108) {
    const int u = t - 96, e = u / 3, c = u % 3;
    float acc = 0.f;
    for (int hp = 0; hp < 96; ++hp)
      acc += o_pt[((size_t)n * 96 + hp) * 3 + c] * Wl1[hp * 4 + e];
    l1_out[(size_t)n * 12 + e * 3 + c] =
        l1_feats[(size_t)n * 12 + e * 3 + c] + acc * rsqrtf(96.0f);
  }
}

// ---------------------------------------------------------------------------
extern "C" void kernel_launch(void* const* d_in, const int* in_sizes, int n_in,
                              void* d_out, int out_size, void* d_ws, size_t ws_size,
                              hipStream_t stream) {
  const float* s     = (const float*)d_in[0];
  const float* z     = (const float*)d_in[1];
  const float* l1f   = (const float*)d_in[2];
  const float* rot   = (const float*)d_in[3];
  const float* trans = (const float*)d_in[4];
  const float* mask  = (const float*)d_in[5];
  const float* Wq    = (const float*)d_in[6];
  const float* bq    = (const float*)d_in[7];
  const float* Wkv   = (const float*)d_in[8];
  const float* bkv   = (const float*)d_in[9];
  const float* Wqp   = (const float*)d_in[10];
  const float* bqp   = (const float*)d_in[11];
  const float* Wkvp  = (const float*)d_in[12];
  const float* bkvp  = (const float*)d_in[13];
  const float* Wb    = (const float*)d_in[14];
  const float* bb    = (const float*)d_in[15];
  const float* Wdz   = (const float*)d_in[16];
  const float* bdz   = (const float*)d_in[17];
  const float* Wout  = (const float*)d_in[18];
  const float* bout  = (const float*)d_in[19];
  const float* Wl1   = (const float*)d_in[20];

  float* ws = (float*)d_ws;
  float* q      = ws;               // 768*192
  float* kv     = q      + 147456;  // 768*384
  float* qp     = kv     + 294912;  // 768*144
  float* kvp    = qp     + 110592;  // 768*432
  float* q_pts  = kvp    + 331776;  // 768*48*3
  float* kv_pts = q_pts  + 110592;  // 768*144*3
  float* qpk    = kv_pts + 331776;  // 768*12*28
  float* kpk    = qpk    + 258048;  // 768*12*28
  float* mQ     = kpk    + 258048;  // 768*12
  float* mK     = mQ     + 9216;    // 768*12
  float* vpk    = mK     + 9216;    // 768*12*40
  float* o_pt   = vpk    + 368640;  // 768*96*3
  float* feats  = o_pt   + 221184;  // 768*960
  float* bmat   = feats  + 737280;  // 768*768*12
  float* att    = bmat   + 7077888; // 12*768*768
  float* pz     = att    + 7077888; // 768*768*32
  // total: 36,218,880 floats (~145 MB)

  float* s_out  = (float*)d_out;
  float* l1_out = s_out + (size_t)N_ * CS_;

  dim3 w32(32);

  // 1) projections from s (K=384)
  gemm_awt_kernel<<<dim3(48, 12), w32, 0, stream>>>(s, Wq,   bq,   nullptr, q,   384, 192, 192);
  gemm_awt_kernel<<<dim3(48, 24), w32, 0, stream>>>(s, Wkv,  bkv,  nullptr, kv,  384, 384, 384);
  gemm_awt_kernel<<<dim3(48,  9), w32, 0, stream>>>(s, Wqp,  bqp,  nullptr, qp,  384, 144, 144);
  gemm_awt_kernel<<<dim3(48, 27), w32, 0, stream>>>(s, Wkvp, bkvp, nullptr, kvp, 384, 432, 432);

  // 2) rotate/translate points, then pack per-head WMMA fragments
  pts_kernel <<<(N_ * 192 + 255) / 256, 256, 0, stream>>>(qp, kvp, rot, trans, q_pts, kv_pts);
  pack_kernel<<<(N_ * H_ * 3 + 255) / 256, 256, 0, stream>>>(q, kv, q_pts, kv_pts,
                                                             qpk, kpk, mQ, mK, vpk);

  // 3) single pass over z (302 MB): b and pair_z fused
  bz_kernel<<<dim3((N_ * N_) / 16), w32, 0, stream>>>(z, Wb, bb, Wdz, bdz, bmat, pz);

  // 4) logits + softmax
  logits_kernel<<<dim3(48, 48, 12), w32, 0, stream>>>(qpk, kpk, mQ, mK, bmat, mask, att);
  softmax_kernel<<<dim3(H_ * N_), w32, 0, stream>>>(att);

  // 5) attention-weighted sums
  ov_kernel   <<<dim3(48, 3, 12), w32, 0, stream>>>(att, vpk, feats, o_pt);
  opair_kernel<<<dim3(768, 2),    w32, 0, stream>>>(att, pz, feats);

  // 6) local-frame points, dists, l1_out; fill feats middle section
  post_kernel<<<dim3(768), dim3(128), 0, stream>>>(o_pt, rot, trans, l1f, Wl1, feats, l1_out);

  // 7) s_out = feats @ Wout^T + bout + s
  gemm_awt_kernel<<<dim3(48, 24), w32, 0, stream>>>(feats, Wout, bout, s, s_out, 960, 384, 384);
}